// GraphTransformerLayer_41755672052078
// MI455X (gfx1250) — compile-verified
//
#include <hip/hip_runtime.h>
#include <hip/hip_bf16.h>
#include <math.h>

// ---------------------------------------------------------------------------
// GraphTransformerLayer on MI455X (gfx1250, wave32)
//   - fp32 WMMA (V_WMMA_F32_16X16X4_F32) for all dense GEMMs (reference is fp32)
//   - 16x64 strip per wave, double-buffered fragments: 5 loads in flight
//     against 4 wmma, compile-time K so tile strides fold into IOFFSET
//   - sparse attention via 4096x4096 adjacency bitmask (2MB, L2-resident),
//     deterministic ascending neighbor order, wave-per-head softmax
// ---------------------------------------------------------------------------

typedef __attribute__((ext_vector_type(2))) float v2f;
typedef __attribute__((ext_vector_type(8))) float v8f;

constexpr int NN = 4096;   // nodes
constexpr int C  = 256;    // channels
constexpr int HH = 8;      // heads
constexpr int DD = 32;     // head dim
constexpr int WPB = 8;     // waves per 256-thread block (attn / LN kernels)
constexpr int GWPB = 4;    // waves per GEMM block (128 threads)
constexpr int NT = 4;      // n-tiles per wave (16x64 output strip)
constexpr int MAXDEG = 1024;

__device__ inline float waveSum(float v) {
  #pragma unroll
  for (int m = 16; m > 0; m >>= 1) v += __shfl_xor(v, m, 32);
  return v;
}

// 16x64 fp32 output strip: c[t] += A(16xK) * B_t(Kx16), B_t[k][n] = W[n0+16t+n][k].
// A: first row of the A-tile (row stride K); B: weight row n0 (row stride K).
// Double-buffered: next k-step's 5 fragments are loaded before this step's wmmas.
template <int K>
__device__ inline void wmma_strip_f32(const float* __restrict__ A,
                                      const float* __restrict__ B,
                                      v8f c[NT]) {
  const int lane = threadIdx.x & 31;
  const int r    = lane & 15;          // A row / B output-col within tile
  const int koff = (lane >> 4) << 1;   // lanes 16-31 carry K+2,K+3
  const float* ap = A + r * K + koff;
  const float* bp = B + r * K + koff;

  #pragma unroll
  for (int t = 0; t < NT; ++t) c[t] = (v8f){0.f,0.f,0.f,0.f,0.f,0.f,0.f,0.f};

  v2f a = *(const v2f*)(ap);
  v2f b[NT];
  #pragma unroll
  for (int t = 0; t < NT; ++t) b[t] = *(const v2f*)(bp + t * 16 * K);

  #pragma unroll 4
  for (int k0 = 0; k0 < K - 4; k0 += 4) {
    // issue next group's loads first (distinct regs -> stays in flight)
    v2f an = *(const v2f*)(ap + k0 + 4);
    v2f bn[NT];
    #pragma unroll
    for (int t = 0; t < NT; ++t)
      bn[t] = *(const v2f*)(bp + t * 16 * K + k0 + 4);
    // current group's matrix ops overlap those loads
    #pragma unroll
    for (int t = 0; t < NT; ++t)
      c[t] = __builtin_amdgcn_wmma_f32_16x16x4_f32(false, a, false, b[t],
                                                   (short)0, c[t], false, false);
    a = an;
    #pragma unroll
    for (int t = 0; t < NT; ++t) b[t] = bn[t];
  }
  // peeled last k-step
  #pragma unroll
  for (int t = 0; t < NT; ++t)
    c[t] = __builtin_amdgcn_wmma_f32_16x16x4_f32(false, a, false, b[t],
                                                 (short)0, c[t], false, false);
}

// Generic GEMM: out[M x Nout] = A[M x K] * W[Nout x K]^T + bias (+ res) (opt GELU)
// grid = (M/16, Nout/64/GWPB), block = 128; one wave per 16x64 strip.
template <int K>
__global__ void k_gemm(const float* __restrict__ A, const float* __restrict__ W,
                       const float* __restrict__ bias, const float* __restrict__ res,
                       float* __restrict__ out, int ldout, int do_gelu) {
  const int wid  = threadIdx.x >> 5;
  const int lane = threadIdx.x & 31;
  const int m0 = blockIdx.x * 16;
  const int n0 = (blockIdx.y * GWPB + wid) * (16 * NT);

  v8f c[NT];
  wmma_strip_f32<K>(A + m0 * K, W + n0 * K, c);

  const int rbase = m0 + ((lane >> 4) << 3);
  #pragma unroll
  for (int t = 0; t < NT; ++t) {
    const int col  = n0 + t * 16 + (lane & 15);
    const float bb = bias[col];
    #pragma unroll
    for (int r = 0; r < 8; ++r) {
      const int row = rbase + r;
      float val = c[t][r] + bb;
      if (res) val += res[row * ldout + col];
      if (do_gelu) val = 0.5f * val * (1.0f + erff(val * 0.70710678118654752440f));
      out[row * ldout + col] = val;
    }
  }
}

// Adjacency bitmask init: zero + self-loop bit per row. idx in [0, N*128).
__global__ void k_adj_init(unsigned* __restrict__ adj) {
  const int idx = blockIdx.x * blockDim.x + threadIdx.x;
  const int row = idx >> 7;
  const int wi  = idx & 127;
  adj[idx] = (wi == (row >> 5)) ? (1u << (row & 31)) : 0u;
}

// Set edge bits: adj[src, dst] = 1 (set semantics -> duplicates collapse).
__global__ void k_adj_edges(unsigned* __restrict__ adj,
                            const int* __restrict__ el, int E) {
  const int e = blockIdx.x * blockDim.x + threadIdx.x;
  if (e >= E) return;
  const int src = el[e];
  const int dst = el[E + e];
  atomicOr(&adj[src * 128 + (dst >> 5)], 1u << (dst & 31));
}

// Sparse masked softmax attention. One block (8 waves) per node; wave = head.
__global__ void k_attn(const unsigned* __restrict__ adj,
                       const float* __restrict__ q, const float* __restrict__ k,
                       const float* __restrict__ v, float* __restrict__ attn) {
  __shared__ unsigned s_nbr[MAXDEG];
  __shared__ float    s_w[HH][MAXDEG];
  __shared__ int      s_cnt[129];

  const int i    = blockIdx.x;
  const int tid  = threadIdx.x;
  const int lane = tid & 31;
  const int h    = tid >> 5;

  // -- deterministic ascending neighbor extraction from bitmask row --
  unsigned word = 0;
  if (tid < 128) {
    word = adj[i * 128 + tid];
    s_cnt[tid] = __popc(word);
  }
  __syncthreads();
  if (tid == 0) {
    int off = 0;
    for (int w = 0; w < 128; ++w) { int c0 = s_cnt[w]; s_cnt[w] = off; off += c0; }
    s_cnt[128] = off;
  }
  __syncthreads();
  if (tid < 128) {
    int pos = s_cnt[tid];
    unsigned wt = word;
    while (wt) {
      const int b = __ffs(wt) - 1;
      if (pos < MAXDEG) s_nbr[pos] = tid * 32 + b;
      ++pos;
      wt &= wt - 1;
    }
  }
  __syncthreads();
  int deg = s_cnt[128];
  if (deg > MAXDEG) deg = MAXDEG;   // statistically impossible (max deg ~45)

  // -- scores: q_i . k_j / sqrt(D), butterfly reduce across the 32-lane dim --
  const float qv = q[i * C + h * DD + lane];
  float mx = -3.4e38f;
  for (int e = 0; e < deg; ++e) {
    const int j = s_nbr[e];
    float d = waveSum(qv * k[j * C + h * DD + lane]);
    const float sc = d * 0.17677669529663688110f;  // 1/sqrt(32)
    if (lane == 0) s_w[h][e] = sc;
    mx = fmaxf(mx, sc);
  }
  // -- softmax weights --
  float sum = 0.f;
  for (int e = 0; e < deg; ++e) {
    const float we = expf(s_w[h][e] - mx);
    sum += we;
    if (lane == 0) s_w[h][e] = we;
  }
  // -- weighted V accumulation (lane = head-dim element) --
  float acc = 0.f;
  for (int e = 0; e < deg; ++e) {
    const int j = s_nbr[e];
    acc += s_w[h][e] * v[j * C + h * DD + lane];
  }
  attn[i * C + h * DD + lane] = acc / sum;
}

// LayerNorm over 256 channels; one wave per row, 8 rows per block.
__global__ void k_ln(const float* __restrict__ in, const float* __restrict__ g,
                     const float* __restrict__ b, float* __restrict__ out) {
  const int wid  = threadIdx.x >> 5;
  const int lane = threadIdx.x & 31;
  const int row  = blockIdx.x * WPB + wid;

  float vals[8];
  float s = 0.f;
  #pragma unroll
  for (int r = 0; r < 8; ++r) {
    vals[r] = in[row * C + r * 32 + lane];
    s += vals[r];
  }
  const float mean = waveSum(s) * (1.0f / 256.0f);
  float vs = 0.f;
  #pragma unroll
  for (int r = 0; r < 8; ++r) {
    const float d = vals[r] - mean;
    vs += d * d;
  }
  const float rstd = rsqrtf(waveSum(vs) * (1.0f / 256.0f) + 1e-5f);
  #pragma unroll
  for (int r = 0; r < 8; ++r) {
    const int col = r * 32 + lane;
    out[row * C + col] = (vals[r] - mean) * rstd * g[col] + b[col];
  }
}

extern "C" void kernel_launch(void* const* d_in, const int* in_sizes, int n_in,
                              void* d_out, int out_size, void* d_ws, size_t ws_size,
                              hipStream_t stream) {
  const float* x    = (const float*)d_in[0];
  const int*   el   = (const int*)  d_in[1];   // [2, E], int32
  const float* Wq   = (const float*)d_in[2];
  const float* bq   = (const float*)d_in[3];
  const float* Wk   = (const float*)d_in[4];
  const float* bk   = (const float*)d_in[5];
  const float* Wv   = (const float*)d_in[6];
  const float* bv   = (const float*)d_in[7];
  const float* Wo   = (const float*)d_in[8];
  const float* bo   = (const float*)d_in[9];
  const float* Wp   = (const float*)d_in[10];
  const float* bp   = (const float*)d_in[11];
  const float* ln1g = (const float*)d_in[12];
  const float* ln1b = (const float*)d_in[13];
  const float* ln2g = (const float*)d_in[14];
  const float* ln2b = (const float*)d_in[15];
  const float* W1   = (const float*)d_in[16];
  const float* b1   = (const float*)d_in[17];
  const float* W2   = (const float*)d_in[18];
  const float* b2   = (const float*)d_in[19];
  float* out = (float*)d_out;

  const int E = in_sizes[1] / 2;

  // workspace layout (bytes): adj 2MB | q 4 | k 4 | v 4 | xp 4 | attn 4 | buf1 4 | h 4
  char* ws = (char*)d_ws;
  unsigned* adj = (unsigned*)ws;                      // NN*128 u32 = 2MB
  float* qb    = (float*)(ws + (2ull << 20));
  float* kb    = qb    + (1u << 20);
  float* vb    = kb    + (1u << 20);
  float* xpb   = vb    + (1u << 20);
  float* attnb = xpb   + (1u << 20);
  float* buf1  = attnb + (1u << 20);                  // out1, later out2
  float* hb    = buf1  + (1u << 20);
  float* tb    = qb;                                  // ffn hidden (N x 512) reuses q+k

  // 1) adjacency bitmask (self-loops folded into init)
  k_adj_init <<<(NN * 128) / 256, 256, 0, stream>>>(adj);
  k_adj_edges<<<(E + 255) / 256, 256, 0, stream>>>(adj, el, E);

  // 2) q/k/v/xp projections: [4096x256] x [256x256]^T  (WMMA f32, 16x64 per wave)
  dim3 gblk(128);
  dim3 g256(NN / 16, C / (16 * NT) / GWPB);          // (256, 1)
  k_gemm<C><<<g256, gblk, 0, stream>>>(x, Wq, bq, nullptr, qb,  C, 0);
  k_gemm<C><<<g256, gblk, 0, stream>>>(x, Wk, bk, nullptr, kb,  C, 0);
  k_gemm<C><<<g256, gblk, 0, stream>>>(x, Wv, bv, nullptr, vb,  C, 0);
  k_gemm<C><<<g256, gblk, 0, stream>>>(x, Wp, bp, nullptr, xpb, C, 0);

  // 3) sparse masked-softmax attention
  k_attn<<<NN, dim3(256), 0, stream>>>(adj, qb, kb, vb, attnb);

  // 4) output projection + xp residual, then LN1 -> h
  k_gemm<C><<<g256, gblk, 0, stream>>>(attnb, Wo, bo, xpb, buf1, C, 0);
  k_ln<<<NN / WPB, dim3(256), 0, stream>>>(buf1, ln1g, ln1b, hb);

  // 5) FFN: gelu(h W1^T + b1) W2^T + b2 + h, then LN2 -> out
  dim3 g512(NN / 16, (2 * C) / (16 * NT) / GWPB);    // (256, 2)
  k_gemm<C><<<g512, gblk, 0, stream>>>(hb, W1, b1, nullptr, tb, 2 * C, 1);
  k_gemm<2 * C><<<g256, gblk, 0, stream>>>(tb, W2, b2, hb, buf1, C, 0);
  k_ln<<<NN / WPB, dim3(256), 0, stream>>>(buf1, ln2g, ln2b, out);
}